// CartesianEquivariantBlock_33036888441238
// MI455X (gfx1250) — compile-verified
//
#include <hip/hip_runtime.h>
#include <cstdint>
#include <cstddef>

// ---------------------------------------------------------------------------
// Problem constants (verified against the Python reference by static_assert)
// ---------------------------------------------------------------------------
constexpr int NU_MAX      = 3;
constexpr int C_OUT_MAX   = 2;
constexpr int NCC         = 1024;   // channels
constexpr int N_SPLITS_C  = 21;     // plan entries
constexpr int N_PARTS_C   = 50;     // (entry, part) GEMMs
constexpr int TOT_C       = 248;    // einsums / weight rows
constexpr int TOTAL_COLS_C= 540;    // sum of 3^r over all parts
constexpr int N_JOBS_C    = 65;     // sum of ceil(3^r/16) N-tiles over parts
constexpr int ASET_COLS   = 120;    // 3+9+27+81

constexpr int POW3[7]     = {1, 3, 9, 27, 81, 243, 729};
constexpr int ASET_OFF[5] = {0, 0, 3, 12, 39};  // column offset of rank-r block

// ---------------------------------------------------------------------------
// Compile-time plan construction (mirrors the Python build_plan exactly,
// including iteration order so weight rows line up)
// ---------------------------------------------------------------------------
struct PartInfo  { int16_t mat; int16_t r; int16_t cols; int16_t col_off; };
struct EntryInfo { int8_t c_out; int8_t nu; int16_t part[3]; };
struct EinsumInfo{ int16_t entry; int8_t c_out; int8_t k; int8_t n; uint8_t var[6]; };

struct Plan {
  EntryInfo  entries[N_SPLITS_C];
  PartInfo   parts[N_PARTS_C];
  EinsumInfo eins[TOT_C];
  int n_entries, n_parts, n_eins, total_cols;
};

struct PairsT    { int8_t a[3]; int8_t b[3]; int8_t k; };
struct MatchList { PairsT m[45]; int count; };

constexpr void gen_matchings(const int* items, int m, PairsT cur, int depth, MatchList& out) {
  if (m == 0) { cur.k = (int8_t)depth; out.m[out.count++] = cur; return; }
  for (int j = 1; j < m; ++j) {
    PairsT nxt = cur;
    nxt.a[depth] = (int8_t)items[0];
    nxt.b[depth] = (int8_t)items[j];
    int rest[6] = {0,0,0,0,0,0};
    int rc = 0;
    for (int t = 1; t < m; ++t) if (t != j) rest[rc++] = items[t];
    gen_matchings(rest, rc, nxt, depth + 1, out);
  }
}

constexpr void gen_pickpairs(int n, int k, MatchList& out) {
  out.count = 0;
  if (k == 0) { PairsT p{}; p.k = 0; out.m[out.count++] = p; return; }
  const int m = 2 * k;
  int idx[6] = {0,0,0,0,0,0};
  for (int i = 0; i < m; ++i) idx[i] = i;
  while (true) {
    int items[6] = {0,0,0,0,0,0};
    for (int i = 0; i < m; ++i) items[i] = idx[i];
    PairsT cur{};
    gen_matchings(items, m, cur, 0, out);
    int i = m - 1;
    while (i >= 0 && idx[i] == n - m + i) --i;
    if (i < 0) break;
    ++idx[i];
    for (int j = i + 1; j < m; ++j) idx[j] = idx[j - 1] + 1;
  }
}

constexpr Plan build_plan() {
  Plan P{};
  int entry_id = 0, part_id = 0, ein_id = 0, col_off = 0;
  for (int c_out = 0; c_out <= C_OUT_MAX; ++c_out) {
    for (int nu = 1; nu <= NU_MAX; ++nu) {
      for (int n = 0; n <= 2 * nu; ++n) {
        if (n - c_out < 0 || (n - c_out) % 2 != 0) continue;
        const int k = (n - c_out) / 2;
        int splits[5][3] = {};
        int ns = 0;
        if (nu == 1) { if (n >= 1) { splits[ns][0] = n; ++ns; } }
        else if (nu == 2) {
          for (int a = 1; a <= n / 2; ++a) { splits[ns][0] = a; splits[ns][1] = n - a; ++ns; }
        } else {
          for (int a = 1; a <= n / 3; ++a)
            for (int b = a; b <= (n - a) / 2; ++b) {
              splits[ns][0] = a; splits[ns][1] = b; splits[ns][2] = n - a - b; ++ns;
            }
        }
        for (int s = 0; s < ns; ++s) {
          EntryInfo& E = P.entries[entry_id];
          E.c_out = (int8_t)c_out; E.nu = (int8_t)nu;
          for (int i = 0; i < nu; ++i) {
            const int r = splits[s][i];
            PartInfo& pp = P.parts[part_id];
            pp.mat     = (int16_t)(entry_id * NU_MAX + i);
            pp.r       = (int16_t)r;
            pp.cols    = (int16_t)POW3[r];
            pp.col_off = (int16_t)col_off;
            col_off += POW3[r];
            E.part[i] = (int16_t)part_id;
            ++part_id;
          }
          MatchList ML{};
          gen_pickpairs(n, k, ML);
          for (int e = 0; e < ML.count; ++e) {
            EinsumInfo& EI = P.eins[ein_id];
            EI.entry = (int16_t)entry_id;
            EI.c_out = (int8_t)c_out; EI.k = (int8_t)k; EI.n = (int8_t)n;
            int var[6] = {-1,-1,-1,-1,-1,-1};
            for (int pr = 0; pr < ML.m[e].k; ++pr) {
              var[(int)ML.m[e].a[pr]] = c_out + pr;
              var[(int)ML.m[e].b[pr]] = c_out + pr;
            }
            int f = 0;
            for (int pos = 0; pos < n; ++pos) if (var[pos] < 0) var[pos] = f++;
            for (int pos = 0; pos < n; ++pos) EI.var[pos] = (uint8_t)var[pos];
            ++ein_id;
          }
          ++entry_id;
        }
      }
    }
  }
  P.n_entries = entry_id; P.n_parts = part_id; P.n_eins = ein_id; P.total_cols = col_off;
  return P;
}

struct Jobs { int16_t part[N_JOBS_C]; int16_t nt[N_JOBS_C]; int count; };

constexpr Jobs build_jobs(const Plan& P) {
  Jobs J{};
  int c = 0;
  for (int p = 0; p < P.n_parts; ++p) {
    const int nt = (P.parts[p].cols + 15) / 16;
    for (int t = 0; t < nt; ++t) { J.part[c] = (int16_t)p; J.nt[c] = (int16_t)t; ++c; }
  }
  J.count = c;
  return J;
}

constexpr Plan PLANC = build_plan();
static_assert(PLANC.n_entries  == N_SPLITS_C,   "entry count mismatch");
static_assert(PLANC.n_parts    == N_PARTS_C,    "part count mismatch");
static_assert(PLANC.n_eins     == TOT_C,        "einsum count mismatch");
static_assert(PLANC.total_cols == TOTAL_COLS_C, "column count mismatch");
constexpr Jobs JOBSC = build_jobs(PLANC);
static_assert(JOBSC.count == N_JOBS_C, "job count mismatch");

__constant__ Plan g_plan = PLANC;
__constant__ Jobs g_jobs = JOBSC;

// ---------------------------------------------------------------------------
// Kernels
// ---------------------------------------------------------------------------
typedef __attribute__((ext_vector_type(2))) float v2f;
typedef __attribute__((ext_vector_type(8))) float v8f;

// (1) a_set[r] = u^{tensor r}, stored as rank-r blocks, row-major (NCC x 3^r)
__global__ __launch_bounds__(256) void build_aset_kernel(const float* __restrict__ u,
                                                         float* __restrict__ aset) {
  const int z = blockIdx.x * 256 + threadIdx.x;
  if (z >= NCC) return;
  const float v[3] = {u[z * 3 + 0], u[z * 3 + 1], u[z * 3 + 2]};
  float* a1 = aset + (size_t)ASET_OFF[1] * NCC + (size_t)z * 3;
  float* a2 = aset + (size_t)ASET_OFF[2] * NCC + (size_t)z * 9;
  float* a3 = aset + (size_t)ASET_OFF[3] * NCC + (size_t)z * 27;
  float* a4 = aset + (size_t)ASET_OFF[4] * NCC + (size_t)z * 81;
  #pragma unroll
  for (int a = 0; a < 3; ++a) {
    a1[a] = v[a];
    #pragma unroll
    for (int b = 0; b < 3; ++b) {
      a2[a * 3 + b] = v[a] * v[b];
      #pragma unroll
      for (int c = 0; c < 3; ++c) {
        a3[a * 9 + b * 3 + c] = v[a] * v[b] * v[c];
        #pragma unroll
        for (int d = 0; d < 3; ++d)
          a4[a * 27 + b * 9 + c * 3 + d] = v[a] * v[b] * v[c] * v[d];
      }
    }
  }
}

// (2) tin_p = mix_p @ aset[r_p]  via V_WMMA_F32_16X16X4_F32.
//     grid.x = N-tile job (part, ntile); each wave owns TWO 16-row M-tiles
//     (rows m0..m0+31) so every B fragment feeds two WMMAs.
//     B slice (1024 x 16) staged in LDS via async-to-LDS copies (ASYNCcnt).
//     No zero padding needed: out-of-range D columns are never stored, and
//     garbage B columns cannot contaminate valid columns. Over-reads stay
//     inside d_ws.
__global__ __launch_bounds__(256) void mix_gemm_kernel(const float* __restrict__ mixes,
                                                       const float* __restrict__ aset,
                                                       float* __restrict__ tins) {
  __shared__ float Bs[NCC * 16];  // 64 KB
  const int job  = blockIdx.x;
  const int part = g_jobs.part[job];
  const int nt   = g_jobs.nt[job];
  const PartInfo pp = g_plan.parts[part];
  const int cols = pp.cols;
  const float* __restrict__ mixm  = mixes + (size_t)pp.mat * NCC * NCC;
  const float* __restrict__ asetb = aset + (size_t)ASET_OFF[pp.r] * NCC;
  const int n0 = nt * 16;

  // --- async B stage: Bs[kk*16 + c] = asetb[kk*cols + n0 + c], 16B chunks ---
  {
    const float* bbase = asetb + n0;   // uniform -> SGPR pair
    for (int i = threadIdx.x; i < NCC * 4; i += 256) {   // 4096 16B chunks
      const int kk = i >> 2, c4 = (i & 3) * 4;
      const unsigned lds_b = (unsigned)(uintptr_t)(&Bs[kk * 16 + c4]); // LDS byte addr
      const unsigned goff  = (unsigned)((kk * cols + c4) * 4);         // byte offset
      asm volatile("global_load_async_to_lds_b128 %0, %1, %2"
                   :: "v"(lds_b), "v"(goff), "s"(bbase) : "memory");
    }
    asm volatile("s_wait_asynccnt 0x0" ::: "memory");
  }
  __syncthreads();

  const int wave = threadIdx.x >> 5;
  const int lane = threadIdx.x & 31;
  const int half = lane >> 4;   // 0: K+{0,1}, 1: K+{2,3}
  const int lid  = lane & 15;   // M within tile for A, N for B/D
  const int m0   = (blockIdx.y * 8 + wave) * 32;   // two 16-row tiles per wave

  // A-frags (16x4 f32): lane(l<16): v0=A[l][k], v1=A[l][k+1]; lane(l>=16): k+2,k+3
  const float* __restrict__ arow0 = mixm + (size_t)(m0 + lid) * NCC + half * 2;
  const float* __restrict__ arow1 = mixm + (size_t)(m0 + 16 + lid) * NCC + half * 2;
  // B-frag (4x16 f32): same K-splitting, N striped across lanes
  const float* __restrict__ bptr = Bs + half * 2 * 16 + lid;

  v8f acc0 = {0.f, 0.f, 0.f, 0.f, 0.f, 0.f, 0.f, 0.f};
  v8f acc1 = {0.f, 0.f, 0.f, 0.f, 0.f, 0.f, 0.f, 0.f};
  #pragma unroll 8
  for (int k = 0; k < NCC; k += 4) {
    v2f b;  b.x  = bptr[k * 16];  b.y  = bptr[k * 16 + 16];
    v2f a0; a0.x = arow0[k];      a0.y = arow0[k + 1];
    v2f a1; a1.x = arow1[k];      a1.y = arow1[k + 1];
    acc0 = __builtin_amdgcn_wmma_f32_16x16x4_f32(false, a0, false, b,
                                                 (short)0, acc0, false, false);
    acc1 = __builtin_amdgcn_wmma_f32_16x16x4_f32(false, a1, false, b,
                                                 (short)0, acc1, false, false);
  }

  // D layout: VGPR v holds row (tile_m0 + v + 8*half), col = n0 + lid
  const int col = n0 + lid;
  if (col < cols) {
    float* __restrict__ op = tins + (size_t)pp.col_off * NCC;
    #pragma unroll
    for (int vtx = 0; vtx < 8; ++vtx) {
      const int row0 = m0 + vtx + half * 8;
      op[(size_t)row0 * cols + col] = acc0[vtx];
    }
    #pragma unroll
    for (int vtx = 0; vtx < 8; ++vtx) {
      const int row1 = m0 + 16 + vtx + half * 8;
      op[(size_t)row1 * cols + col] = acc1[vtx];
    }
  }
}

// (3) zero the output (harness poisons it)
__global__ __launch_bounds__(256) void zero_out_kernel(float* __restrict__ out, int n) {
  const int i = blockIdx.x * 256 + threadIdx.x;
  if (i < n) out[i] = 0.0f;
}

// (4) table-driven einsum contraction + weight combine.
//     One thread per (einsum t, channel z); t uniform within a block.
__global__ __launch_bounds__(256) void contract_kernel(const float* __restrict__ tins,
                                                       const float* __restrict__ weights,
                                                       float* __restrict__ out) {
  const int gid = blockIdx.x * 256 + threadIdx.x;
  const int z = gid & (NCC - 1);
  const int t = gid >> 10;
  if (t >= TOT_C) return;

  const EinsumInfo E = g_plan.eins[t];
  const EntryInfo  S = g_plan.entries[E.entry];

  const float* tp[3] = {nullptr, nullptr, nullptr};
  int rr[3] = {0, 0, 0};
  for (int i = 0; i < S.nu; ++i) {
    const PartInfo pp = g_plan.parts[S.part[i]];
    tp[i] = tins + (size_t)pp.col_off * NCC + (size_t)z * pp.cols;
    rr[i] = pp.r;
  }
  const float w = weights[(size_t)t * NCC + z];
  const int c = E.c_out, kk = E.k;
  const int p3c = POW3[c], p3k = POW3[kk];
  const int out_base = (c == 0) ? z : (c == 1 ? (NCC + z * 3) : (NCC * 4 + z * 9));

  int vals[6] = {0, 0, 0, 0, 0, 0};
  for (int f = 0; f < p3c; ++f) {
    int tf = f;
    for (int j = c - 1; j >= 0; --j) { vals[j] = tf % 3; tf /= 3; }
    float sum = 0.f;
    for (int a = 0; a < p3k; ++a) {
      int ta = a;
      for (int j = 0; j < kk; ++j) { vals[c + j] = ta % 3; ta /= 3; }
      float prod = 1.f;
      int pb = 0;
      for (int i = 0; i < S.nu; ++i) {
        int idx = 0;
        for (int j = 0; j < rr[i]; ++j) idx = idx * 3 + vals[E.var[pb + j]];
        prod *= tp[i][idx];
        pb += rr[i];
      }
      sum += prod;
    }
    atomicAdd(&out[out_base + f], w * sum);
  }
}

// ---------------------------------------------------------------------------
// Launch
// ---------------------------------------------------------------------------
extern "C" void kernel_launch(void* const* d_in, const int* in_sizes, int n_in,
                              void* d_out, int out_size, void* d_ws, size_t ws_size,
                              hipStream_t stream) {
  (void)in_sizes; (void)n_in; (void)ws_size;
  const float* u       = (const float*)d_in[0];
  const float* mixes   = (const float*)d_in[1];
  const float* weights = (const float*)d_in[2];
  float* out  = (float*)d_out;
  float* aset = (float*)d_ws;                       // 120*1024 f32  (~0.5 MB)
  float* tins = aset + (size_t)ASET_COLS * NCC;     // 540*1024 f32  (~2.2 MB)

  build_aset_kernel<<<dim3((NCC + 255) / 256), dim3(256), 0, stream>>>(u, aset);
  // 65 N-tile jobs x (64 M-tiles / (8 waves * 2 tiles per wave)) = (65, 4)
  mix_gemm_kernel<<<dim3(N_JOBS_C, 4), dim3(256), 0, stream>>>(mixes, aset, tins);
  zero_out_kernel<<<dim3((out_size + 255) / 256), dim3(256), 0, stream>>>(out, out_size);
  contract_kernel<<<dim3((TOT_C * NCC) / 256), dim3(256), 0, stream>>>(tins, weights, out);
}